// TopKAttentionPoolingWithNHeadsMultiModalProjector_87540023427683
// MI455X (gfx1250) — compile-verified
//
#include <hip/hip_runtime.h>

#define ENC_H   1536
#define TXT_H   4096
#define N_PATCH 4096
#define TOP_K   64

typedef __attribute__((ext_vector_type(2))) float v2f;
typedef __attribute__((ext_vector_type(8))) float v8f;

// Workspace layout (float offsets):
#define WS_V      0        // v[1536]   folded score vector
#define WS_C      2048     // c         folded score bias (scalar)
#define WS_SCORES 4096     // scores[4096]
#define WS_KEEP   8192     // keep[64] (int)
#define WS_XK     16384    // Xk[64*1536] gathered kept rows
// total: (16384 + 64*1536)*4 = 458,752 bytes of d_ws

// ---------------------------------------------------------------- init v = 0
__global__ void tk_init_v(float* ws) {
    int i = blockIdx.x * blockDim.x + threadIdx.x;
    if (i < ENC_H) ws[WS_V + i] = 0.0f;
}

// -------------------------------------------- v[e] = sum_t Ws[t]*Wp[t][e]
__global__ void tk_fold_v(const float* __restrict__ Wp,
                          const float* __restrict__ Ws, float* ws) {
    int e  = blockIdx.x * 256 + threadIdx.x;   // gridDim.x = 6
    int t0 = blockIdx.y * 256;                 // gridDim.y = 16
    float acc = 0.0f;
#pragma unroll 4
    for (int t = t0; t < t0 + 256; ++t)
        acc = fmaf(Ws[t], Wp[(size_t)t * ENC_H + e], acc);
    atomicAdd(&ws[WS_V + e], acc);
}

// -------------------------------------------- c = dot(Ws, b_proj) + b_score
__global__ void tk_fold_c(const float* __restrict__ Ws,
                          const float* __restrict__ bp,
                          const float* __restrict__ bs, float* ws) {
    __shared__ float red[256];
    float acc = 0.0f;
    for (int t = threadIdx.x; t < TXT_H; t += 256)
        acc = fmaf(Ws[t], bp[t], acc);
    red[threadIdx.x] = acc;
    __syncthreads();
    for (int s = 128; s > 0; s >>= 1) {
        if ((int)threadIdx.x < s) red[threadIdx.x] += red[threadIdx.x + s];
        __syncthreads();
    }
    if (threadIdx.x == 0) ws[WS_C] = red[0] + bs[0];
}

// -------------------------------------------- scores[n] = dot(x[n], v) + c
__global__ void tk_scores(const float* __restrict__ x, float* ws) {
    int wave = threadIdx.x >> 5;               // 8 waves/block
    int lane = threadIdx.x & 31;
    int row  = blockIdx.x * 8 + wave;          // gridDim.x = 512
    const float* v  = ws + WS_V;
    const float* xr = x + (size_t)row * ENC_H;
    float acc = 0.0f;
    for (int e = lane; e < ENC_H; e += 32)
        acc = fmaf(xr[e], v[e], acc);
    for (int off = 16; off > 0; off >>= 1)
        acc += __shfl_xor(acc, off, 32);
    if (lane == 0) ws[WS_SCORES + row] = acc + ws[WS_C];
}

// ------------- top-64 by score; stable tie-break (lower index wins), output
// ------------- ascending indices == jnp.sort(argsort(-attn)[:64])
__global__ void __launch_bounds__(1024) tk_topk(float* ws) {
    __shared__ float         s[N_PATCH];
    __shared__ unsigned char kept[N_PATCH];
    int tid = threadIdx.x;
    for (int i = tid; i < N_PATCH; i += 1024) s[i] = ws[WS_SCORES + i];
    __syncthreads();
    for (int i = tid; i < N_PATCH; i += 1024) {
        float sv = s[i];
        int rank = 0;
        for (int m = 0; m < N_PATCH; ++m) {
            float sm = s[m];
            rank += (sm > sv) || ((sm == sv) && (m < i));
        }
        kept[i] = (rank < TOP_K) ? 1 : 0;
    }
    __syncthreads();
    if (tid == 0) {
        int* keep = (int*)(ws + WS_KEEP);
        int pos = 0;
        for (int i = 0; i < N_PATCH && pos < TOP_K; ++i)
            if (kept[i]) keep[pos++] = i;
    }
}

// -------------------------------------------- gather kept rows -> Xk (64x1536)
__global__ void tk_gather(const float* __restrict__ x, float* ws) {
    const int* keep = (const int*)(ws + WS_KEEP);
    int row = keep[blockIdx.x];                // gridDim.x = 64
    const float* src = x + (size_t)row * ENC_H;
    float* dst = ws + WS_XK + (size_t)blockIdx.x * ENC_H;
    for (int e = threadIdx.x; e < ENC_H; e += 256) dst[e] = src[e];
}

// ---- out(64x4096) = Xk(64x1536) @ Wp^T(1536x4096) + b_proj, via fp32 WMMA.
// Each wave: one 16-wide N strip, all 4 M-tiles. K stepped by 4 (16x16x4).
// A layout: lane m=lane&15, k = 2*(lane>>4)+{0,1}  -> float2 from Xk row m.
// B layout: lane n=lane&15, k = 2*(lane>>4)+{0,1}  -> float2 from Wp row n
//           (B = Wp^T is column-major, so k is contiguous within a Wp row).
// C/D layout: VGPR j -> row j + 8*(lane>>4), col = lane&15.
__global__ void __launch_bounds__(256) tk_wmma_gemm(
        const float* __restrict__ Wp, const float* __restrict__ bp,
        const float* ws, float* __restrict__ out) {
    const float* Xk = ws + WS_XK;
    int wave  = blockIdx.x * 8 + (threadIdx.x >> 5);   // 32 blocks -> 256 waves
    int lane  = threadIdx.x & 31;
    int nbase = wave * 16;
    int g     = lane >> 4;     // k-pair group
    int n     = lane & 15;     // A row (m) / B col (n) for this lane

    v8f acc0 = {}, acc1 = {}, acc2 = {}, acc3 = {};
    const float* Brow  = Wp + (size_t)(nbase + n) * ENC_H + 2 * g;
    const float* Arow0 = Xk + (size_t)(0 * 16 + n) * ENC_H + 2 * g;
    const float* Arow1 = Xk + (size_t)(1 * 16 + n) * ENC_H + 2 * g;
    const float* Arow2 = Xk + (size_t)(2 * 16 + n) * ENC_H + 2 * g;
    const float* Arow3 = Xk + (size_t)(3 * 16 + n) * ENC_H + 2 * g;

    for (int kb = 0; kb < ENC_H; kb += 4) {
        __builtin_prefetch(Brow + kb + 128, 0, 1);     // stream-ahead on Wp
        v2f b  = *(const v2f*)(Brow  + kb);
        v2f a0 = *(const v2f*)(Arow0 + kb);
        v2f a1 = *(const v2f*)(Arow1 + kb);
        v2f a2 = *(const v2f*)(Arow2 + kb);
        v2f a3 = *(const v2f*)(Arow3 + kb);
        acc0 = __builtin_amdgcn_wmma_f32_16x16x4_f32(false, a0, false, b, (short)0, acc0, false, false);
        acc1 = __builtin_amdgcn_wmma_f32_16x16x4_f32(false, a1, false, b, (short)0, acc1, false, false);
        acc2 = __builtin_amdgcn_wmma_f32_16x16x4_f32(false, a2, false, b, (short)0, acc2, false, false);
        acc3 = __builtin_amdgcn_wmma_f32_16x16x4_f32(false, a3, false, b, (short)0, acc3, false, false);
    }

    float bias = bp[nbase + n];
    int col  = nbase + n;
    int rofs = 8 * g;
#pragma unroll
    for (int j = 0; j < 8; ++j) {
        out[(size_t)( 0 + j + rofs) * TXT_H + col] = acc0[j] + bias;
        out[(size_t)(16 + j + rofs) * TXT_H + col] = acc1[j] + bias;
        out[(size_t)(32 + j + rofs) * TXT_H + col] = acc2[j] + bias;
        out[(size_t)(48 + j + rofs) * TXT_H + col] = acc3[j] + bias;
    }
}

extern "C" void kernel_launch(void* const* d_in, const int* in_sizes, int n_in,
                              void* d_out, int out_size, void* d_ws, size_t ws_size,
                              hipStream_t stream) {
    const float* x  = (const float*)d_in[0];   // image_features (1,4096,1536)
    const float* Wp = (const float*)d_in[1];   // W_proj (4096,1536)
    const float* bp = (const float*)d_in[2];   // b_proj (4096)
    const float* Ws = (const float*)d_in[3];   // W_score (1,4096)
    const float* bs = (const float*)d_in[4];   // b_score (1)
    float* ws  = (float*)d_ws;
    float* out = (float*)d_out;

    hipLaunchKernelGGL(tk_init_v,    dim3(6),          dim3(256),  0, stream, ws);
    hipLaunchKernelGGL(tk_fold_v,    dim3(6, 16),      dim3(256),  0, stream, Wp, Ws, ws);
    hipLaunchKernelGGL(tk_fold_c,    dim3(1),          dim3(256),  0, stream, Ws, bp, bs, ws);
    hipLaunchKernelGGL(tk_scores,    dim3(N_PATCH/8),  dim3(256),  0, stream, x, ws);
    hipLaunchKernelGGL(tk_topk,      dim3(1),          dim3(1024), 0, stream, ws);
    hipLaunchKernelGGL(tk_gather,    dim3(TOP_K),      dim3(256),  0, stream, x, ws);
    hipLaunchKernelGGL(tk_wmma_gemm, dim3(32),         dim3(256),  0, stream, Wp, bp, ws, out);
}